// GNN_14886356648486
// MI455X (gfx1250) — compile-verified
//
#include <hip/hip_runtime.h>

typedef __attribute__((ext_vector_type(2))) float v2f;
typedef __attribute__((ext_vector_type(8))) float v8f;

// ---------------- degree / CSR construction ----------------

__global__ __launch_bounds__(256) void k_zero_i32(int* __restrict__ p, int n) {
  int i = blockIdx.x * blockDim.x + threadIdx.x;
  if (i < n) p[i] = 0;
}

__global__ __launch_bounds__(256) void k_count(const int* __restrict__ dst,
                                               int* __restrict__ counts, int e) {
  int i = blockIdx.x * blockDim.x + threadIdx.x;
  if (i < e) atomicAdd(&counts[dst[i]], 1);
}

__global__ __launch_bounds__(256) void k_dinv(const int* __restrict__ counts,
                                              float* __restrict__ dinv, int n) {
  int i = blockIdx.x * blockDim.x + threadIdx.x;
  if (i < n) dinv[i] = rsqrtf((float)counts[i] + 1.0f);  // deg = in-degree + 1 (self loop)
}

__global__ __launch_bounds__(512) void k_scan_local(const int* __restrict__ counts,
                                                    int* __restrict__ rowptr,
                                                    int* __restrict__ partials, int n) {
  __shared__ int sm[512];
  int tid = threadIdx.x;
  int i = blockIdx.x * 512 + tid;
  int v = (i < n) ? counts[i] : 0;
  sm[tid] = v;
  __syncthreads();
  for (int off = 1; off < 512; off <<= 1) {   // Hillis-Steele inclusive scan
    int a = sm[tid];
    int b = (tid >= off) ? sm[tid - off] : 0;
    __syncthreads();
    sm[tid] = a + b;
    __syncthreads();
  }
  if (i < n) rowptr[i] = sm[tid] - v;         // exclusive within block
  if (tid == 511) partials[blockIdx.x] = sm[511];
}

__global__ void k_scan_partials(int* __restrict__ partials, int nb) {
  if (blockIdx.x == 0 && threadIdx.x == 0) {
    int run = 0;
    for (int b = 0; b < nb; ++b) { int t = partials[b]; partials[b] = run; run += t; }
  }
}

__global__ __launch_bounds__(512) void k_scan_add(int* __restrict__ rowptr,
                                                  int* __restrict__ cursor,
                                                  const int* __restrict__ partials,
                                                  int n, int total) {
  int i = blockIdx.x * 512 + threadIdx.x;
  if (i < n) {
    int v = rowptr[i] + partials[blockIdx.x];
    rowptr[i] = v;
    cursor[i] = v;
  }
  if (i == 0) rowptr[n] = total;
}

__global__ __launch_bounds__(256) void k_scatter(const int* __restrict__ src,
                                                 const int* __restrict__ dst,
                                                 int* __restrict__ cursor,
                                                 int* __restrict__ srcs, int e) {
  int i = blockIdx.x * blockDim.x + threadIdx.x;
  if (i < e) {
    int pos = atomicAdd(&cursor[dst[i]], 1);
    srcs[pos] = src[i];
  }
}

// ---------------- fp32 WMMA GEMM: H[n,DOUT] = X[n,K] @ W[K,DOUT] ----------------
// One wave computes a 16-row strip across all DOUT columns using
// V_WMMA_F32_16X16X4_F32 (fp32 in / fp32 accumulate, preserves reference precision).
// A 16x4 layout: lane m=lane&15 is the row; half-wave kh=lane>>4 selects K pair
// {0,1} (VGPR0/1) vs {2,3}. B 4x16 symmetric. C/D: VGPR j -> M=j+8*kh, N=lane&15.

template <int K, int DOUT>
__global__ __launch_bounds__(256) void k_gemm_wmma(const float* __restrict__ X,
                                                   const float* __restrict__ W,
                                                   float* __restrict__ H, int n) {
  int gw   = (blockIdx.x * blockDim.x + threadIdx.x) >> 5;   // global wave id
  int lane = threadIdx.x & 31;
  int row0 = gw * 16;
  if (row0 >= n) return;                                     // wave-uniform: EXEC stays all-1s

  constexpr int NT = DOUT / 16;
  v8f acc[NT] = {};
  const int m  = lane & 15;
  const int kh = lane >> 4;

  for (int k0 = 0; k0 < K; k0 += 4) {
    const float* xr = X + (size_t)(row0 + m) * K + (k0 + 2 * kh);
    v2f a;
    a.x = xr[0];
    a.y = xr[1];
#pragma unroll
    for (int t = 0; t < NT; ++t) {
      const float* wr = W + (size_t)(k0 + 2 * kh) * DOUT + t * 16 + m;
      v2f b;
      b.x = wr[0];
      b.y = wr[DOUT];
      acc[t] = __builtin_amdgcn_wmma_f32_16x16x4_f32(
          /*neg_a=*/false, a, /*neg_b=*/false, b,
          /*c_mod=*/(short)0, acc[t], /*reuse_a=*/false, /*reuse_b=*/false);
    }
  }

#pragma unroll
  for (int t = 0; t < NT; ++t) {
#pragma unroll
    for (int j = 0; j < 8; ++j) {
      H[(size_t)(row0 + j + 8 * kh) * DOUT + (t * 16 + m)] = acc[t][j];
    }
  }
}

// ---------------- CSR gather aggregation ----------------
// out[i,:] = sum_{e in row i} dinv[src_e]*dinv[i]*h[src_e,:] + dinv[i]^2*h[i,:] + b
// One wave per node; lanes stride feature dim (coalesced). h rows live in L2 (192MB).

template <int D, bool RELU>
__global__ __launch_bounds__(256) void k_aggregate(const float* __restrict__ h,
                                                   const int* __restrict__ rowptr,
                                                   const int* __restrict__ srcs,
                                                   const float* __restrict__ dinv,
                                                   const float* __restrict__ bias,
                                                   float* __restrict__ out, int n) {
  int wave = (blockIdx.x * blockDim.x + threadIdx.x) >> 5;
  int lane = threadIdx.x & 31;
  if (wave >= n) return;

  int beg = rowptr[wave];
  int end = rowptr[wave + 1];
  float di = dinv[wave];

  constexpr int NC = (D + 31) / 32;
  float acc[NC];
#pragma unroll
  for (int j = 0; j < NC; ++j) {
    int c = lane + 32 * j;
    acc[j] = (c < D) ? di * di * h[(size_t)wave * D + c] : 0.0f;  // self-loop term
  }

  for (int e = beg; e < end; ++e) {
    int s = srcs[e];
    float coef = di * dinv[s];
#pragma unroll
    for (int j = 0; j < NC; ++j) {
      int c = lane + 32 * j;
      if (c < D) acc[j] = fmaf(coef, h[(size_t)s * D + c], acc[j]);
    }
  }

#pragma unroll
  for (int j = 0; j < NC; ++j) {
    int c = lane + 32 * j;
    if (c < D) {
      float v = acc[j] + bias[c];
      if (RELU) v = fmaxf(v, 0.0f);
      out[(size_t)wave * D + c] = v;
    }
  }
}

// ---------------- driver ----------------

extern "C" void kernel_launch(void* const* d_in, const int* in_sizes, int n_in,
                              void* d_out, int out_size, void* d_ws, size_t ws_size,
                              hipStream_t stream) {
  const float* feat = (const float*)d_in[0];
  const int*   ei   = (const int*)d_in[1];
  const float* W1   = (const float*)d_in[2];
  const float* b1   = (const float*)d_in[3];
  const float* W2   = (const float*)d_in[4];
  const float* b2   = (const float*)d_in[5];
  const float* W3   = (const float*)d_in[6];
  const float* b3   = (const float*)d_in[7];

  constexpr int IN = 8, HID = 64, OUT = 112;
  const int n = in_sizes[0] / IN;   // 100000
  const int e = in_sizes[1] / 2;    // 1600000
  const int* src = ei;
  const int* dst = ei + e;

  // workspace carve-out (~78.5 MB)
  char* ws = (char*)d_ws;
  size_t off = 0;
  auto carve = [&](size_t bytes) -> void* {
    void* p = ws + off;
    off = (off + bytes + 255) & ~(size_t)255;
    return p;
  };
  int*   counts   = (int*)carve((size_t)n * 4);
  int*   rowptr   = (int*)carve(((size_t)n + 1) * 4);
  int*   cursor   = (int*)carve((size_t)n * 4);
  float* dinv     = (float*)carve((size_t)n * 4);
  int*   partials = (int*)carve(4096);
  int*   srcs     = (int*)carve((size_t)e * 4);
  float* h        = (float*)carve((size_t)n * OUT * 4);  // widest layer
  float* x1       = (float*)carve((size_t)n * HID * 4);  // reused for both activations
  (void)ws_size; (void)n_in; (void)out_size;

  const int nb_n256 = (n + 255) / 256;
  const int nb_e256 = (e + 255) / 256;
  const int nb_scan = (n + 511) / 512;

  // ---- CSR build (once, reused by all 3 layers) ----
  k_zero_i32<<<nb_n256, 256, 0, stream>>>(counts, n);
  k_count<<<nb_e256, 256, 0, stream>>>(dst, counts, e);
  k_dinv<<<nb_n256, 256, 0, stream>>>(counts, dinv, n);
  k_scan_local<<<nb_scan, 512, 0, stream>>>(counts, rowptr, partials, n);
  k_scan_partials<<<1, 32, 0, stream>>>(partials, nb_scan);
  k_scan_add<<<nb_scan, 512, 0, stream>>>(rowptr, cursor, partials, n, e);
  k_scatter<<<nb_e256, 256, 0, stream>>>(src, dst, cursor, srcs, e);

  const int gemm_waves  = (n + 15) / 16;
  const int gemm_blocks = (gemm_waves + 7) / 8;   // 8 waves / 256-thread block
  const int agg_blocks  = (n + 7) / 8;            // 8 nodes / block (1 wave each)

  // ---- layer 1: [n,8]@[8,64] -> agg+ReLU ----
  k_gemm_wmma<8, 64><<<gemm_blocks, 256, 0, stream>>>(feat, W1, h, n);
  k_aggregate<64, true><<<agg_blocks, 256, 0, stream>>>(h, rowptr, srcs, dinv, b1, x1, n);

  // ---- layer 2: [n,64]@[64,64] -> agg+ReLU (x1 safely overwritten after gemm) ----
  k_gemm_wmma<64, 64><<<gemm_blocks, 256, 0, stream>>>(x1, W2, h, n);
  k_aggregate<64, true><<<agg_blocks, 256, 0, stream>>>(h, rowptr, srcs, dinv, b2, x1, n);

  // ---- layer 3: [n,64]@[64,112] -> agg (no ReLU) -> d_out ----
  k_gemm_wmma<64, 112><<<gemm_blocks, 256, 0, stream>>>(x1, W3, h, n);
  k_aggregate<112, false><<<agg_blocks, 256, 0, stream>>>(h, rowptr, srcs, dinv, b3,
                                                          (float*)d_out, n);
}